// DepthwiseDilationLayer_78168404787232
// MI455X (gfx1250) — compile-verified
//
#include <hip/hip_runtime.h>

typedef float v4f __attribute__((ext_vector_type(4)));

#define TILE_H  32
#define RS      232                 // LDS row stride in floats (16B aligned: 232*4 = 928 = 58*16)
#define NROWS   (TILE_H + 2)        // 34 rows (1 halo above, 1 below)
#define LDS_FLT (NROWS * RS)        // 7888 floats = 31552 bytes

// out[b,c,h,w] = max_{r,s in -1..1} xpad[b,c,h+r,w+s] + k[c, 3*(r+1)+(s+1)]
__global__ __launch_bounds__(256, 4)
void dilate3x3_async_kernel(const float* __restrict__ x,
                            const float* __restrict__ kern,
                            float* __restrict__ out)
{
    __shared__ __align__(16) float s[LDS_FLT];

    const int tid   = threadIdx.x;
    const int tile  = blockIdx.x;        // 0..6  (7 * 32 = 224 rows)
    const int plane = blockIdx.y;        // 0..511 (b*64 + c)
    const int c     = plane & 63;
    const int gr0   = tile * TILE_H;     // first output row of this tile

    const float* xp = x   + (size_t)plane * (224 * 224);
    float*       op = out + (size_t)plane * (224 * 224);

    // ---- 1) zero the halo (pad columns every row; pad top/bottom row at image edge) ----
    // Data cols live at LDS cols [4..227]; left-pad zero at col 3, right-pad zero at col 228.
    {
        const v4f z = {0.f, 0.f, 0.f, 0.f};
        for (int i = tid; i < NROWS * 2; i += 256) {
            int r = i >> 1;
            int colbase = (i & 1) ? 228 : 0;     // zero cols 0..3 and 228..231
            *(v4f*)&s[r * RS + colbase] = z;
        }
        if (gr0 == 0) {                           // top image edge -> LDS row 0 is zero
            for (int q = tid; q < 56; q += 256)
                *(v4f*)&s[0 * RS + 4 + q * 4] = z;
        }
        if (gr0 + TILE_H >= 224) {                // bottom image edge -> LDS row 33 is zero
            for (int q = tid; q < 56; q += 256)
                *(v4f*)&s[(NROWS - 1) * RS + 4 + q * 4] = z;
        }
    }
    __syncthreads();

    // ---- 2) async DMA the valid input rows into LDS (b128 per lane) ----
    {
        int gr_first  = gr0 - 1;
        int lds_first = 0;
        if (gr_first < 0) { gr_first = 0; lds_first = 1; }
        int gr_last = gr0 + TILE_H;               // inclusive halo row below
        if (gr_last > 223) gr_last = 223;
        const int nrows = gr_last - gr_first + 1; // 33 or 34
        const int total = nrows * 56;             // 56 float4 per 224-wide row

        const unsigned long long sbase = (unsigned long long)(uintptr_t)xp;
        for (int l = tid; l < total; l += 256) {
            int r = l / 56;
            int q = l - r * 56;
            unsigned voff = (unsigned)((gr_first + r) * 896 + q * 16);  // global byte offset
            unsigned ldsoff =
                (unsigned)(uintptr_t)(&s[(lds_first + r) * RS + 4 + q * 4]); // LDS byte addr
            // GVS addressing: mem = SADDR(64) + VADDR(32) ; VDST = LDS byte address
            asm volatile("global_load_async_to_lds_b128 %0, %1, %2"
                         :
                         : "v"(ldsoff), "v"(voff), "s"(sbase)
                         : "memory");
        }
        asm volatile("s_wait_asynccnt 0" ::: "memory");
    }
    __syncthreads();

    // ---- 3) per-channel 3x3 additive SE ----
    const float* kc = kern + c * 9;
    float kk[9];
    #pragma unroll
    for (int i = 0; i < 9; ++i) kk[i] = kc[i];

    // ---- 4) each thread computes 7 float4 quads (1792 = 7 * 256 exactly) ----
    #pragma unroll
    for (int it = 0; it < 7; ++it) {
        const int idx = tid + it * 256;           // 0..1791
        const int oh  = idx / 56;                 // 0..31
        const int ow4 = idx - oh * 56;            // 0..55

        v4f acc = {-__builtin_inff(), -__builtin_inff(),
                   -__builtin_inff(), -__builtin_inff()};

        #pragma unroll
        for (int r = 0; r < 3; ++r) {
            const float* row = &s[(oh + r) * RS + 3 + ow4 * 4];
            float a  = row[0];                       // input col 4*ow4 - 1
            v4f   m  = *(const v4f*)(row + 1);       // input cols 4*ow4 .. 4*ow4+3 (16B aligned)
            float b5 = row[5];                       // input col 4*ow4 + 4
            const float ka = kk[3 * r + 0];
            const float kb = kk[3 * r + 1];
            const float kd = kk[3 * r + 2];
            acc.x = fmaxf(acc.x, fmaxf(fmaxf(a   + ka, m.x + kb), m.y + kd));
            acc.y = fmaxf(acc.y, fmaxf(fmaxf(m.x + ka, m.y + kb), m.z + kd));
            acc.z = fmaxf(acc.z, fmaxf(fmaxf(m.y + ka, m.z + kb), m.w + kd));
            acc.w = fmaxf(acc.w, fmaxf(fmaxf(m.z + ka, m.w + kb), b5  + kd));
        }

        // write-once output: stream past near caches (keep input halo lines in L2)
        __builtin_nontemporal_store(acc, (v4f*)&op[(gr0 + oh) * 224 + ow4 * 4]);
    }
}

extern "C" void kernel_launch(void* const* d_in, const int* in_sizes, int n_in,
                              void* d_out, int out_size, void* d_ws, size_t ws_size,
                              hipStream_t stream) {
    (void)in_sizes; (void)n_in; (void)d_ws; (void)ws_size; (void)out_size;
    const float* x    = (const float*)d_in[0];   // [8,64,224,224]
    const float* kern = (const float*)d_in[1];   // [1,64,9,1,1]
    float* out        = (float*)d_out;           // [8,64,224,224]

    dim3 grid(7, 8 * 64, 1);   // 7 row-tiles x 512 planes
    dim3 block(256, 1, 1);
    dilate3x3_async_kernel<<<grid, block, 0, stream>>>(x, kern, out);
}